// model_SagPoolGCN_54769422958883
// MI455X (gfx1250) — compile-verified
//
#include <hip/hip_runtime.h>
#include <hip/hip_bf16.h>
#include <math.h>

#define N_PER_GRAPH 1024
#define K_KEEP      512
#define HIDDEN      128
#define OUT_CH      8
#define DEG         16
#define G_GRAPHS    64
#define NODES       (G_GRAPHS * N_PER_GRAPH)      // 65536
#define NP          (G_GRAPHS * K_KEEP)           // 32768 kept nodes
#define NEDGE       (NODES * DEG)                 // 1048576
#define EPG         (N_PER_GRAPH * DEG)           // 16384 edges/graph

typedef __attribute__((ext_vector_type(2))) float v2f;
typedef __attribute__((ext_vector_type(8))) float v8f;

// ---------------- utility ----------------
__global__ void zero_f32(float* __restrict__ p, int n) {
    int i = blockIdx.x * blockDim.x + threadIdx.x;
    if (i < n) p[i] = 0.0f;
}

// ---------------- stage A: original-graph degree + dinv ----------------
__global__ void deg_kernel(const int* __restrict__ col, const float* __restrict__ w,
                           float* __restrict__ deg) {
    int e = blockIdx.x * blockDim.x + threadIdx.x;
    if (e < NEDGE) atomicAdd(&deg[col[e]], w[e]);
}

__global__ void dinv_kernel(float* __restrict__ deg_dinv, int n) {
    int i = blockIdx.x * blockDim.x + threadIdx.x;
    if (i < n) deg_dinv[i] = rsqrtf(deg_dinv[i] + 1.0f);   // +1 = self loop; always > 0
}

// ---------------- stage B: SAG score (GCNConv 1->1) ----------------
__global__ void score_scatter(const int* __restrict__ row, const int* __restrict__ col,
                              const float* __restrict__ w, const float* __restrict__ x,
                              const float* __restrict__ dinv, float* __restrict__ sagg) {
    int e = blockIdx.x * blockDim.x + threadIdx.x;
    if (e < NEDGE) {
        int r = row[e], c = col[e];
        atomicAdd(&sagg[c], dinv[r] * w[e] * dinv[c] * x[r]);
    }
}

__global__ void score_finalize(const float* __restrict__ sagg, const float* __restrict__ x,
                               const float* __restrict__ dinv, const float* __restrict__ sagW,
                               const float* __restrict__ sagB, float* __restrict__ score) {
    int i = blockIdx.x * blockDim.x + threadIdx.x;
    if (i < NODES) {
        float s = sagg[i] + dinv[i] * dinv[i] * x[i];       // self loop
        score[i] = s * sagW[0] + sagB[0];
    }
}

// ---------------- stage C: top-512 per graph via exact rank (stable tie-break) ----
__global__ __launch_bounds__(256) void topk_kernel(const float* __restrict__ score,
                                                   int* __restrict__ newid,
                                                   int* __restrict__ perm) {
    __shared__ float s[N_PER_GRAPH];
    int g = blockIdx.x;
    for (int i = threadIdx.x; i < N_PER_GRAPH; i += blockDim.x)
        s[i] = score[g * N_PER_GRAPH + i];
    __syncthreads();
    for (int i = threadIdx.x; i < N_PER_GRAPH; i += blockDim.x) {
        float si = s[i];
        int rank = 0;
        for (int j = 0; j < N_PER_GRAPH; ++j) {
            float sj = s[j];
            rank += (sj > si) || (sj == si && j < i);       // matches lax.top_k stability
        }
        int node = g * N_PER_GRAPH + i;
        if (rank < K_KEEP) {
            int nid = g * K_KEEP + rank;                    // descending-score order
            newid[node] = nid;
            perm[nid] = node;
        } else {
            newid[node] = -1;
        }
    }
}

// ---------------- stage D: gate by tanh(score), relu ----------------
__global__ void gate_kernel(const float* __restrict__ x, const float* __restrict__ score,
                            const int* __restrict__ perm, float* __restrict__ h0) {
    int p = blockIdx.x * blockDim.x + threadIdx.x;
    if (p < NP) {
        int node = perm[p];
        float v = x[node] * tanhf(score[node]);
        h0[p] = v > 0.0f ? v : 0.0f;
    }
}

// ---------------- stage E: relabel edges, pooled-graph degree ----------------
__global__ void relabel_kernel(const int* __restrict__ row, const int* __restrict__ col,
                               const float* __restrict__ w, const int* __restrict__ newid,
                               int* __restrict__ r2, int* __restrict__ c2,
                               float* __restrict__ deg2) {
    int e = blockIdx.x * blockDim.x + threadIdx.x;
    if (e < NEDGE) {
        int nr = newid[row[e]], nc = newid[col[e]];
        bool valid = (nr >= 0) && (nc >= 0);
        r2[e] = valid ? nr : -1;
        c2[e] = valid ? nc : -1;
        if (valid) atomicAdd(&deg2[nc], w[e]);
    }
}

__global__ void dinv2_kernel(float* __restrict__ deg2_dinv2, float* __restrict__ selfc, int n) {
    int i = blockIdx.x * blockDim.x + threadIdx.x;
    if (i < n) {
        float d = rsqrtf(deg2_dinv2[i] + 1.0f);
        deg2_dinv2[i] = d;
        selfc[i] = d * d;                                   // self-loop coefficient
    }
}

__global__ void coef_kernel(const int* __restrict__ r2, const int* __restrict__ c2,
                            const float* __restrict__ w, const float* __restrict__ dinv2,
                            float* __restrict__ coef) {
    int e = blockIdx.x * blockDim.x + threadIdx.x;
    if (e < NEDGE) {
        int r = r2[e];
        coef[e] = (r >= 0) ? dinv2[r] * w[e] * dinv2[c2[e]] : 0.0f;
    }
}

// ---------------- stage F: conv1 (rank-1 -> scalar scatter + outer product) ----
__global__ void l1_scatter(const int* __restrict__ r2, const int* __restrict__ c2,
                           const float* __restrict__ coef, const float* __restrict__ h0,
                           float* __restrict__ s1) {
    int e = blockIdx.x * blockDim.x + threadIdx.x;
    if (e < NEDGE) {
        int r = r2[e];
        if (r >= 0) atomicAdd(&s1[c2[e]], coef[e] * h0[r]);
    }
}

__global__ void l1_expand(const float* __restrict__ s1, const float* __restrict__ selfc,
                          const float* __restrict__ h0, const float* __restrict__ W1,
                          const float* __restrict__ b1, float* __restrict__ H) {
    int idx = blockIdx.x * blockDim.x + threadIdx.x;      // NP*128 threads
    if (idx < NP * HIDDEN) {
        int p = idx >> 7, c = idx & 127;
        float s = s1[p] + selfc[p] * h0[p];
        float v = s * W1[c] + b1[c];
        H[idx] = v > 0.0f ? v : 0.0f;
    }
}

// ---------------- GEMM: XW = H(32768x128) @ W(128x128), f32 WMMA -------------
// One wave computes a 16x16 C tile with V_WMMA_F32_16X16X4_F32 (K stepped by 4).
// A 16x4 f32 frag: lanes 0-15 -> M=lane, K={k0,k0+1}; lanes 16-31 -> K={k0+2,k0+3}.
// B 4x16 f32 frag mirrors with N across lanes. C/D: 8 VGPRs per ISA layout.
__global__ __launch_bounds__(256) void gemm_kernel(const float* __restrict__ H,
                                                   const float* __restrict__ W,
                                                   float* __restrict__ XW) {
    int lane = threadIdx.x & 31;
    int wave = threadIdx.x >> 5;          // 8 waves -> 8 N tiles (covers all 128 cols)
    int m0 = blockIdx.x * 16;             // 2048 blocks cover 32768 rows
    int n0 = wave * 16;
    int half = lane >> 4;                 // 0: K pair {0,1}; 1: K pair {2,3}
    int l = lane & 15;
    const float* hrow = H + (size_t)(m0 + l) * HIDDEN;
    v8f c = {};
    for (int k0 = 0; k0 < HIDDEN; k0 += 4) {
        int ka = k0 + 2 * half;
        v2f a, b;
        a.x = hrow[ka];
        a.y = hrow[ka + 1];
        b.x = W[ka * HIDDEN + n0 + l];
        b.y = W[(ka + 1) * HIDDEN + n0 + l];
        c = __builtin_amdgcn_wmma_f32_16x16x4_f32(false, a, false, b,
                                                  (short)0, c, false, false);
    }
    int mbase = m0 + half * 8;
#pragma unroll
    for (int v = 0; v < 8; ++v)
        XW[(size_t)(mbase + v) * HIDDEN + n0 + l] = c[v];
}

// ---------------- edge scatter with per-graph LDS accumulation ---------------
// One workgroup per graph; output tile 512x128 f32 = 256 KB of the 320 KB WGP LDS.
// tile initialized with the self-loop term, edges accumulated via ds_add_f32,
// epilogue adds bias + relu and writes Hout.
__global__ __launch_bounds__(256) void scatter_layer(const float* __restrict__ XW,
                                                     const int* __restrict__ r2,
                                                     const int* __restrict__ c2,
                                                     const float* __restrict__ coef,
                                                     const float* __restrict__ selfc,
                                                     const float* __restrict__ bias,
                                                     float* __restrict__ Hout) {
    extern __shared__ float tile[];                        // K_KEEP * HIDDEN
    int g = blockIdx.x;
    int base_node = g * K_KEEP;
    for (int i = threadIdx.x; i < K_KEEP * HIDDEN; i += blockDim.x) {
        int r = i >> 7;
        tile[i] = selfc[base_node + r] * XW[(size_t)(base_node + r) * HIDDEN + (i & 127)];
    }
    __syncthreads();
    int lane = threadIdx.x & 31;
    int wave = threadIdx.x >> 5;
    int ebase = g * EPG;
    for (int ei = wave; ei < EPG; ei += 8) {               // wave-per-edge (uniform branch)
        int e = ebase + ei;
        int r = r2[e];
        if (r < 0) continue;
        int lc = c2[e] - base_node;
        float f = coef[e];
        const float* src = XW + (size_t)r * HIDDEN;
#pragma unroll
        for (int j = 0; j < 4; ++j) {
            int cc = lane + 32 * j;                        // bank-conflict-free ds_add
            atomicAdd(&tile[lc * HIDDEN + cc], f * src[cc]);
        }
    }
    __syncthreads();
    for (int i = threadIdx.x; i < K_KEEP * HIDDEN; i += blockDim.x) {
        int r = i >> 7, cc = i & 127;
        float v = tile[i] + bias[cc];
        Hout[(size_t)(base_node + r) * HIDDEN + cc] = v > 0.0f ? v : 0.0f;
    }
}

// ---------------- mean pool + head ----------------
__global__ void pool_kernel(const float* __restrict__ H, float* __restrict__ pooled) {
    int g = blockIdx.x, c = threadIdx.x;                   // 128 threads
    float s = 0.0f;
    for (int r = 0; r < K_KEEP; ++r)
        s += H[(size_t)(g * K_KEEP + r) * HIDDEN + c];
    pooled[g * HIDDEN + c] = s / (float)K_KEEP;
}

__global__ void head_kernel(const float* __restrict__ pooled, const float* __restrict__ Wout,
                            const float* __restrict__ bout, float* __restrict__ out) {
    __shared__ float l[OUT_CH];
    int g = blockIdx.x, o = threadIdx.x;                   // 8 threads
    float acc = bout[o];
    for (int c = 0; c < HIDDEN; ++c)
        acc += pooled[g * HIDDEN + c] * Wout[c * OUT_CH + o];
    l[o] = acc;
    __syncthreads();
    float m = -INFINITY;
    for (int i = 0; i < OUT_CH; ++i) m = fmaxf(m, l[i]);
    float s = 0.0f;
    for (int i = 0; i < OUT_CH; ++i) s += expf(l[i] - m);
    out[g * OUT_CH + o] = acc - m - logf(s);
}

// ---------------- launch ----------------
extern "C" void kernel_launch(void* const* d_in, const int* in_sizes, int n_in,
                              void* d_out, int out_size, void* d_ws, size_t ws_size,
                              hipStream_t stream) {
    (void)in_sizes; (void)n_in; (void)out_size; (void)ws_size;

    const float* x    = (const float*)d_in[0];
    const int*   eidx = (const int*)d_in[1];               // [2, E] flat
    const float* ew   = (const float*)d_in[2];
    // d_in[3] = batch (implied by layout; unused)
    const float* sagW = (const float*)d_in[4];
    const float* sagB = (const float*)d_in[5];
    const float* W1   = (const float*)d_in[6];
    const float* b1   = (const float*)d_in[7];
    const float* W2   = (const float*)d_in[8];
    const float* b2   = (const float*)d_in[9];
    const float* W3   = (const float*)d_in[10];
    const float* b3   = (const float*)d_in[11];
    const float* Wout = (const float*)d_in[12];
    const float* bout = (const float*)d_in[13];
    float* out = (float*)d_out;

    const int* row = eidx;
    const int* col = eidx + NEDGE;

    // workspace layout (256B aligned)
    char* base = (char*)d_ws;
    size_t off = 0;
    auto alloc = [&](size_t bytes) {
        void* p = base + off;
        off += (bytes + 255) & ~(size_t)255;
        return p;
    };
    float* dinv   = (float*)alloc(NODES * 4);      // deg then dinv (in place)
    float* sagg   = (float*)alloc(NODES * 4);
    float* score  = (float*)alloc(NODES * 4);
    int*   newid  = (int*)  alloc(NODES * 4);
    int*   perm   = (int*)  alloc(NP * 4);
    float* h0     = (float*)alloc(NP * 4);
    float* dinv2  = (float*)alloc(NP * 4);         // deg2 then dinv2 (in place)
    float* selfc  = (float*)alloc(NP * 4);
    float* s1     = (float*)alloc(NP * 4);
    int*   r2     = (int*)  alloc((size_t)NEDGE * 4);
    int*   c2     = (int*)  alloc((size_t)NEDGE * 4);
    float* coef   = (float*)alloc((size_t)NEDGE * 4);
    float* bufA   = (float*)alloc((size_t)NP * HIDDEN * 4);
    float* bufB   = (float*)alloc((size_t)NP * HIDDEN * 4);
    float* pooled = (float*)alloc(G_GRAPHS * HIDDEN * 4);

    const int T = 256;
    const int EB = (NEDGE + T - 1) / T;                    // 4096
    const int NB = (NODES + T - 1) / T;                    // 256
    const int PB = (NP + T - 1) / T;                       // 128
    const size_t lds_bytes = (size_t)K_KEEP * HIDDEN * 4;  // 256 KB

    hipFuncSetAttribute((const void*)scatter_layer,
                        hipFuncAttributeMaxDynamicSharedMemorySize, (int)lds_bytes);

    // A: degree + dinv of original graph
    zero_f32<<<NB, T, 0, stream>>>(dinv, NODES);
    deg_kernel<<<EB, T, 0, stream>>>(col, ew, dinv);
    dinv_kernel<<<NB, T, 0, stream>>>(dinv, NODES);

    // B: SAG score
    zero_f32<<<NB, T, 0, stream>>>(sagg, NODES);
    score_scatter<<<EB, T, 0, stream>>>(row, col, ew, x, dinv, sagg);
    score_finalize<<<NB, T, 0, stream>>>(sagg, x, dinv, sagW, sagB, score);

    // C: per-graph top-512
    topk_kernel<<<G_GRAPHS, T, 0, stream>>>(score, newid, perm);

    // D: gate + relu
    gate_kernel<<<PB, T, 0, stream>>>(x, score, perm, h0);

    // E: relabel + pooled-graph normalization
    zero_f32<<<PB, T, 0, stream>>>(dinv2, NP);
    relabel_kernel<<<EB, T, 0, stream>>>(row, col, ew, newid, r2, c2, dinv2);
    dinv2_kernel<<<PB, T, 0, stream>>>(dinv2, selfc, NP);
    coef_kernel<<<EB, T, 0, stream>>>(r2, c2, ew, dinv2, coef);

    // F: conv1 (rank-1)
    zero_f32<<<PB, T, 0, stream>>>(s1, NP);
    l1_scatter<<<EB, T, 0, stream>>>(r2, c2, coef, h0, s1);
    l1_expand<<<(NP * HIDDEN + T - 1) / T, T, 0, stream>>>(s1, selfc, h0, W1, b1, bufA);

    // conv2: GEMM (WMMA f32) then LDS-tile edge scatter
    gemm_kernel<<<NP / 16, T, 0, stream>>>(bufA, W2, bufB);
    scatter_layer<<<G_GRAPHS, T, lds_bytes, stream>>>(bufB, r2, c2, coef, selfc, b2, bufA);

    // conv3
    gemm_kernel<<<NP / 16, T, 0, stream>>>(bufA, W3, bufB);
    scatter_layer<<<G_GRAPHS, T, lds_bytes, stream>>>(bufB, r2, c2, coef, selfc, b3, bufA);

    // pool + head
    pool_kernel<<<G_GRAPHS, HIDDEN, 0, stream>>>(bufA, pooled);
    head_kernel<<<G_GRAPHS, OUT_CH, 0, stream>>>(pooled, Wout, bout, out);
}